// HybridKernel_65481071400246
// MI455X (gfx1250) — compile-verified
//
#include <hip/hip_runtime.h>
#include <math.h>

#define D 32
#define GAMMA 1.0f
#define ALPHA 0.5f
#define EPSN 1e-8f

typedef float v2f __attribute__((ext_vector_type(2)));
typedef float v8f __attribute__((ext_vector_type(8)));

// ---------------- Stage 1: per-column mean and 1/(std+eps) (std unbiased) ----
__global__ void hk_stats_kernel(const float* __restrict__ x,
                                const float* __restrict__ y,
                                float* __restrict__ stats, // [4*D]: mux,rsx,muy,rsy
                                int N, int M) {
  __shared__ float s_sum[256];
  __shared__ float s_sq[256];
  const int mat = blockIdx.x / D;
  const int c   = blockIdx.x % D;
  const float* src = mat ? y : x;
  const int rows = mat ? M : N;
  float sum = 0.f, sq = 0.f;
  for (int r = threadIdx.x; r < rows; r += 256) {
    float v = src[(size_t)r * D + c];
    sum += v; sq += v * v;
  }
  s_sum[threadIdx.x] = sum; s_sq[threadIdx.x] = sq;
  __syncthreads();
  for (int o = 128; o > 0; o >>= 1) {
    if (threadIdx.x < o) {
      s_sum[threadIdx.x] += s_sum[threadIdx.x + o];
      s_sq[threadIdx.x]  += s_sq[threadIdx.x + o];
    }
    __syncthreads();
  }
  if (threadIdx.x == 0) {
    float mu  = s_sum[0] / (float)rows;
    float var = (s_sq[0] - (float)rows * mu * mu) / (float)(rows - 1);
    var = var > 0.f ? var : 0.f;
    float rs = 1.f / (sqrtf(var) + EPSN);
    stats[mat * 2 * D + c]     = mu;
    stats[mat * 2 * D + D + c] = rs;
  }
}

// ---------------- Stage 2: normalize, row sq-norms (wave32 shfl), trig tables
__global__ void hk_prep_kernel(const float* __restrict__ x, const float* __restrict__ y,
                               const float* __restrict__ stats,
                               float* __restrict__ xn, float* __restrict__ yn,
                               float* __restrict__ nx, float* __restrict__ ny,
                               float2* __restrict__ csx, float2* __restrict__ csy,
                               int N, int M) {
  const int lane = threadIdx.x & 31;          // lane == feature dim (D==32==wave32)
  const int gw   = (blockIdx.x * blockDim.x + threadIdx.x) >> 5;
  if (gw >= N + M) return;
  const int mat = gw >= N;
  const int row = mat ? gw - N : gw;
  const float* src = mat ? y : x;
  float v  = src[(size_t)row * D + lane];
  float mu = stats[mat * 2 * D + lane];
  float rs = stats[mat * 2 * D + D + lane];
  float nv = (v - mu) * rs;
  (mat ? yn : xn)[(size_t)row * D + lane] = nv;
  float2 cs; cs.x = cosf(0.5f * v); cs.y = sinf(0.5f * v);
  (mat ? csy : csx)[(size_t)row * D + lane] = cs;
  float sq = nv * nv;
  #pragma unroll
  for (int o = 16; o > 0; o >>= 1) sq += __shfl_xor(sq, o, 32);
  if (lane == 0) (mat ? ny : nx)[row] = sq;
}

// ---------------- Stage 3: fused hybrid kernel -------------------------------
// Block = 256 threads (8 waves). Block tile = 128 rows x 16 cols.
// Wave w owns the 16x16 tile at rows i0+16w.
__global__ __launch_bounds__(256) void hk_hybrid_kernel(
    const float* __restrict__ xn, const float* __restrict__ yn,
    const float* __restrict__ nx, const float* __restrict__ ny,
    const float2* __restrict__ csx, const float2* __restrict__ csy,
    float* __restrict__ out, int N, int M) {
  __shared__ float2 sCX[128][33];   // (cos,sin)(x/2), padded stride -> no bank conflicts
  __shared__ float2 sCY[16][33];

  const int tid  = threadIdx.x;
  const int lane = tid & 31;
  const int w    = tid >> 5;
  const int j0   = blockIdx.x * 16;
  const int i0   = blockIdx.y * 128;

  // Stage trig tiles into LDS.
  #pragma unroll
  for (int t = 0; t < 16; ++t) {
    int idx = tid + t * 256;
    int r = idx >> 5, d = idx & 31;
    sCX[r][d] = csx[(size_t)(i0 + r) * D + d];
  }
  {
    int r = tid >> 5, d = tid & 31;
    sCY[r][d] = csy[(size_t)(j0 + r) * D + d];
    int idx = tid + 256; r = idx >> 5; d = idx & 31;
    sCY[r][d] = csy[(size_t)(j0 + r) * D + d];
  }
  __syncthreads();

  const int h = lane >> 4;        // half-wave select
  const int n = lane & 15;        // N index (and A's M index) for this lane

  // ---- classical: dot(xn_i, yn_j) via 8x V_WMMA_F32_16X16X4_F32 (K = 32) ----
  v8f acc = {};
  const float* aptr = xn + (size_t)(i0 + 16 * w + n) * D + 2 * h;
  const float* bptr = yn + (size_t)(j0 + n) * D + 2 * h;
  #pragma unroll
  for (int k = 0; k < 8; ++k) {
    v2f a = *(const v2f*)(aptr + 4 * k);
    v2f b = *(const v2f*)(bptr + 4 * k);
    acc = __builtin_amdgcn_wmma_f32_16x16x4_f32(false, a, false, b,
                                                (short)0, acc, false, false);
  }

  // ---- quantum: prod_d (cx*cy + sx*sy), squared at the end -------------------
  float p[8];
  #pragma unroll
  for (int v = 0; v < 8; ++v) p[v] = 1.f;
  const int rbase = 16 * w + 8 * h;   // block-local row base for this lane's 8 slots
  #pragma unroll 4
  for (int d = 0; d < D; ++d) {
    float2 cy = sCY[n][d];
    #pragma unroll
    for (int v = 0; v < 8; ++v) {
      float2 cx = sCX[rbase + v][d];
      p[v] *= cx.x * cy.x + cx.y * cy.y;
    }
  }

  // ---- epilogue: combine, matching the C/D WMMA layout (M = v + 8h) ----------
  const int j = j0 + n;
  const float nyj = ny[j];
  #pragma unroll
  for (int v = 0; v < 8; ++v) {
    int i = i0 + rbase + v;
    float sq  = nx[i] + nyj - 2.f * acc[v];     // ||xn_i - yn_j||^2
    float cls = expf(-GAMMA * sq);
    float q   = p[v] * p[v];                    // prod cos^2
    out[(size_t)i * M + j] = (1.f - ALPHA) * cls + ALPHA * q;
  }
}

// ---------------- launcher ----------------------------------------------------
extern "C" void kernel_launch(void* const* d_in, const int* in_sizes, int n_in,
                              void* d_out, int out_size, void* d_ws, size_t ws_size,
                              hipStream_t stream) {
  const float* x = (const float*)d_in[0];
  const float* y = (const float*)d_in[1];
  float* out = (float*)d_out;
  const int N = in_sizes[0] / D;   // 2048
  const int M = in_sizes[1] / D;   // 2048

  // Workspace layout (floats). Total ~1.6 MB.
  float* ws    = (float*)d_ws;
  float* stats = ws;                               // 4*D
  float* xn    = stats + 4 * D;                    // N*D
  float* yn    = xn + (size_t)N * D;               // M*D
  float* nx    = yn + (size_t)M * D;               // N
  float* ny    = nx + N;                           // M
  float2* csx  = (float2*)(ny + M);                // N*D float2
  float2* csy  = csx + (size_t)N * D;              // M*D float2

  hk_stats_kernel<<<2 * D, 256, 0, stream>>>(x, y, stats, N, M);

  int waves = N + M;
  hk_prep_kernel<<<(waves * 32 + 255) / 256, 256, 0, stream>>>(
      x, y, stats, xn, yn, nx, ny, csx, csy, N, M);

  dim3 grid(M / 16, N / 128);
  hk_hybrid_kernel<<<grid, 256, 0, stream>>>(xn, yn, nx, ny, csx, csy, out, N, M);
}